// MedianPool2d_45999099740684
// MI455X (gfx1250) — compile-verified
//
#include <hip/hip_runtime.h>

// 3x3 zero-padded median filter over (16,3,512,512) fp32.
// Bandwidth-bound stencil: ~100 MB min traffic -> ~4.3 us floor @ 23.3 TB/s.
// LDS-tiled, wave32-friendly, coalesced stores, async global->LDS (gfx1250).

#define TILE 32
#define HALO 1
#define TW   (TILE + 2 * HALO)   // 34: tile + halo
#define TP   (TW + 2)            // 36: padded LDS row stride (conflict-free)

static constexpr int Himg = 512;
static constexpr int Wimg = 512;
static constexpr int NPLANES = 16 * 3;   // B*C

// Address-space pointer typedefs for the async-copy builtin
// (probe showed param0 is a non-const AS1 int*).
typedef __attribute__((address_space(1))) int  as1_int;
typedef __attribute__((address_space(3))) int  as3_int;

#define SORT2(a, b)                       \
  {                                       \
    float _t = fminf((a), (b));           \
    (b) = fmaxf((a), (b));                \
    (a) = _t;                             \
  }

__global__ __launch_bounds__(256) void median3x3_kernel(
    const float* __restrict__ in, float* __restrict__ out) {
  __shared__ float tile[TW * TP];

  const int tid = threadIdx.x;
  const int bx = blockIdx.x;   // tile col
  const int by = blockIdx.y;   // tile row
  const int plane = blockIdx.z;
  const float* __restrict__ src = in + (size_t)plane * (Himg * Wimg);
  float* __restrict__ dst = out + (size_t)plane * (Himg * Wimg);

  // ---- Stage 1: fill 34x34 LDS tile (zero-padded halo) ----
#if __has_builtin(__builtin_amdgcn_global_load_async_to_lds_b32)
  // gfx1250 async path: zero-fill, then EXEC-masked async copy of in-bounds
  // elements (tracked by ASYNCcnt, no VGPR round trip).
  for (int i = tid; i < TW * TP; i += 256) tile[i] = 0.0f;
  __syncthreads();
  for (int i = tid; i < TW * TW; i += 256) {
    const int ly = i / TW;
    const int lx = i - ly * TW;
    const int gy = by * TILE + ly - HALO;
    const int gx = bx * TILE + lx - HALO;
    if ((unsigned)gy < (unsigned)Himg && (unsigned)gx < (unsigned)Wimg) {
      __builtin_amdgcn_global_load_async_to_lds_b32(
          (as1_int*)(src + gy * Wimg + gx),
          (as3_int*)&tile[ly * TP + lx],
          /*offset=*/0, /*cpol=*/0);
    }
  }
#if __has_builtin(__builtin_amdgcn_s_wait_asynccnt)
  __builtin_amdgcn_s_wait_asynccnt(0);
#else
  asm volatile("s_wait_asynccnt 0" ::: "memory");
#endif
  __syncthreads();
#else
  // Fallback: global_load_b32 -> ds_store_b32
  for (int i = tid; i < TW * TW; i += 256) {
    const int ly = i / TW;
    const int lx = i - ly * TW;
    const int gy = by * TILE + ly - HALO;
    const int gx = bx * TILE + lx - HALO;
    float v = 0.0f;
    if ((unsigned)gy < (unsigned)Himg && (unsigned)gx < (unsigned)Wimg)
      v = src[gy * Wimg + gx];
    tile[ly * TP + lx] = v;
  }
  __syncthreads();
#endif

  // ---- Stage 2: each thread computes 4 vertically adjacent outputs ----
  const int tx = tid & 31;        // 0..31 output col within tile
  const int ty = tid >> 5;        // 0..7
  const int r0 = ty * 4;          // output row base within tile (0..28)

  // 6 rows x 3 cols register patch covers the four 3x3 windows.
  float v[6][3];
#pragma unroll
  for (int rr = 0; rr < 6; ++rr) {
#pragma unroll
    for (int cc = 0; cc < 3; ++cc) {
      v[rr][cc] = tile[(r0 + rr) * TP + tx + cc];
    }
  }

  const int gx = bx * TILE + tx;
#pragma unroll
  for (int k = 0; k < 4; ++k) {
    float p0 = v[k][0],     p1 = v[k][1],     p2 = v[k][2];
    float p3 = v[k + 1][0], p4 = v[k + 1][1], p5 = v[k + 1][2];
    float p6 = v[k + 2][0], p7 = v[k + 2][1], p8 = v[k + 2][2];

    // Classic 19-exchange median-of-9 network (result in p4).
    SORT2(p1, p2); SORT2(p4, p5); SORT2(p7, p8);
    SORT2(p0, p1); SORT2(p3, p4); SORT2(p6, p7);
    SORT2(p1, p2); SORT2(p4, p5); SORT2(p7, p8);
    SORT2(p0, p3); SORT2(p5, p8); SORT2(p4, p7);
    SORT2(p3, p6); SORT2(p1, p4); SORT2(p2, p5);
    SORT2(p4, p7); SORT2(p4, p2); SORT2(p6, p4);
    SORT2(p4, p2);

    const int gy = by * TILE + r0 + k;
    dst[gy * Wimg + gx] = p4;   // lane-consecutive in x -> coalesced
  }
}

extern "C" void kernel_launch(void* const* d_in, const int* in_sizes, int n_in,
                              void* d_out, int out_size, void* d_ws, size_t ws_size,
                              hipStream_t stream) {
  (void)in_sizes; (void)n_in; (void)d_ws; (void)ws_size; (void)out_size;
  const float* x = (const float*)d_in[0];
  float* out = (float*)d_out;

  dim3 grid(Wimg / TILE, Himg / TILE, NPLANES);  // (16,16,48)
  dim3 block(256, 1, 1);                         // 8 wave32s
  median3x3_kernel<<<grid, block, 0, stream>>>(x, out);
}